// LoRALinear_10883447128289
// MI455X (gfx1250) — compile-verified
//
#include <hip/hip_runtime.h>

// CDNA5 (gfx1250) fused LoRA forward:  out = (x @ A) @ B
//   x: [8192, 4096] f32, A: [4096, 16] f32, B: [16, 4096] f32
// Factored path is ~2.1 GFLOP (vs 274 GFLOP in the reference) and purely
// HBM-bound: 134 MB in + 134 MB out => ~11.5 us at 23.3 TB/s.
// Matrix math uses V_WMMA_F32_16X16X4_F32 (wave32).
//
// Per workgroup (256 thr = 8 waves): own 16 rows of x.
//   Stage 1: waves split K=4096 (512 each), partial 16x16 T tiles via WMMA,
//            deterministic fixed-order LDS reduction (no float atomics).
//   Stage 2: waves split N=4096 (512 cols each), out = T @ B as 32 tiles,
//            K=16 done as 4 chained 16x16x4 WMMAs. T never hits global.

typedef __attribute__((ext_vector_type(2))) float v2f;
typedef __attribute__((ext_vector_type(8))) float v8f;

#define D_IN   4096
#define D_OUT  4096
#define RANK   16

__global__ __launch_bounds__(256) void lora_fused_wmma(
    const float* __restrict__ x,    // [M, D_IN]
    const float* __restrict__ Am,   // [D_IN, RANK]
    const float* __restrict__ Bm,   // [RANK, D_OUT]
    float* __restrict__ out)        // [M, D_OUT]
{
    const int tid   = threadIdx.x;
    const int w     = tid >> 5;       // wave id 0..7
    const int lane  = tid & 31;
    const int half  = lane >> 4;      // 0: lanes 0-15, 1: lanes 16-31
    const int l16   = lane & 15;
    const int khalf = half * 2;       // K offset owned by this lane half
    const int m0    = blockIdx.x * 16;

    __shared__ float red[8 * 256];    // per-wave partial 16x16 tiles
    __shared__ float Tl[256];         // final T tile (16 rows x 16 rank)

    // ---------------- Stage 1: T_tile = x[m0:m0+16, :] @ A ----------------
    // Wave w covers K range [w*512, w*512+512). A-frag: lane holds row
    // m = l16, K pair {k+khalf, k+khalf+1} -> one contiguous b64 load.
    // Constant trip count (128) keeps loop control scalar: no v_cmp/EXEC
    // masking around the WMMAs.
    const float* xp = x + (size_t)(m0 + l16) * D_IN + (w * 512 + khalf);
    const float* ap = Am + (size_t)(w * 512 + khalf) * RANK + l16;

    v8f acc = {0.f, 0.f, 0.f, 0.f, 0.f, 0.f, 0.f, 0.f};
#pragma unroll 4
    for (int it = 0; it < 128; ++it) {
        v2f a = *(const v2f*)(xp);                          // global_load_b64
        v2f b;
        b.x = ap[0];                                        // A row k+khalf
        b.y = ap[RANK];                                     // A row k+khalf+1
        acc = __builtin_amdgcn_wmma_f32_16x16x4_f32(
            false, a, false, b, (short)0, acc, false, false);
        xp += 4;
        ap += 4 * RANK;
    }

    // Deterministic cross-wave reduction of the 8 partial 16x16 tiles.
    // C/D layout: VGPR i, lane L -> element (M = i + 8*half, N = l16).
#pragma unroll
    for (int i = 0; i < 8; ++i)
        red[w * 256 + (i + 8 * half) * 16 + l16] = acc[i];
    __syncthreads();

    {
        float s = 0.f;
#pragma unroll
        for (int wv = 0; wv < 8; ++wv) s += red[wv * 256 + tid];
        Tl[tid] = s;                  // Tl[m*16 + r]
    }
    __syncthreads();

    // ---------------- Stage 2: out[m0:m0+16, :] = T_tile @ B ----------------
    // A-frags come from LDS (contiguous b64 per lane), reused for all tiles.
    v2f af[4];
#pragma unroll
    for (int j = 0; j < 4; ++j) {
        af[j].x = Tl[l16 * 16 + 4 * j + khalf];
        af[j].y = Tl[l16 * 16 + 4 * j + khalf + 1];
    }

    // Wave w covers output columns [w*512, w*512+512) as 32 tiles of 16.
    // B (256 KB) is fully L2-resident, so re-reads per workgroup are cheap.
    for (int t = 0; t < 32; ++t) {
        const int n0 = w * 512 + t * 16;
        v8f c = {0.f, 0.f, 0.f, 0.f, 0.f, 0.f, 0.f, 0.f};
#pragma unroll
        for (int j = 0; j < 4; ++j) {
            const float* bp = Bm + (size_t)(4 * j + khalf) * D_OUT + n0 + l16;
            v2f b;
            b.x = bp[0];              // B row 4j+khalf   (64B coalesced/half)
            b.y = bp[D_OUT];          // B row 4j+khalf+1
            c = __builtin_amdgcn_wmma_f32_16x16x4_f32(
                false, af[j], false, b, (short)0, c, false, false);
        }
        // Store: VGPR i -> out[m0 + i + 8*half, n0 + l16]; 64B coalesced runs.
        float* op = out + (size_t)(m0 + 8 * half) * D_OUT + n0 + l16;
#pragma unroll
        for (int i = 0; i < 8; ++i)
            op[(size_t)i * D_OUT] = c[i];
    }
}

extern "C" void kernel_launch(void* const* d_in, const int* in_sizes, int n_in,
                              void* d_out, int out_size, void* d_ws, size_t ws_size,
                              hipStream_t stream) {
    const float* x  = (const float*)d_in[0];   // [B*S, D_IN] flat
    const float* Am = (const float*)d_in[1];   // [D_IN, RANK]
    const float* Bm = (const float*)d_in[2];   // [RANK, D_OUT]
    float* out = (float*)d_out;

    const int M = in_sizes[0] / D_IN;          // 8192
    dim3 grid(M / 16), block(256);
    lora_fused_wmma<<<grid, block, 0, stream>>>(x, Am, Bm, out);
}